// SS2D_6158983102553
// MI455X (gfx1250) — compile-verified
//
#include <hip/hip_runtime.h>
#include <cstdint>

typedef __attribute__((ext_vector_type(2))) float v2f;
typedef __attribute__((ext_vector_type(8))) float v8f;

#if defined(__has_builtin)
#  if __has_builtin(__builtin_amdgcn_wmma_f32_16x16x4_f32)
#    define HAVE_WMMA_F32 1
#  endif
#endif
#ifndef HAVE_WMMA_F32
#  define HAVE_WMMA_F32 0
#endif

// ---- problem constants ----
#define B_SZ   2
#define H_SZ   64
#define W_SZ   64
#define L_SZ   4096        // H*W
#define DM_SZ  96
#define DE_SZ  192
#define N_ST   16
#define R_SZ   6
#define K_DIR  4
#define CPROJ  38          // R + 2N
#define XDBL_STRIDE 40     // padded row: [0:6)=dt rows, [8:24)=B, [24:40)=C

// ---- workspace layout (float offsets) ----
#define OFF_XC0  0                                            // (B, DE, L)  conv input (xp, channel-major)
#define OFF_ZS   (OFF_XC0  + B_SZ*DE_SZ*L_SZ)                 // (B*L, DE)   silu(z)
#define OFF_XC1  (OFF_ZS   + (size_t)B_SZ*L_SZ*DE_SZ)         // (B, DE, L)  conv output
#define OFF_XDBL (OFF_XC1  + (size_t)B_SZ*DE_SZ*L_SZ)         // (B,K,L,40)  x_dbl, scan order, l-major
#define OFF_YS   (OFF_XDBL + (size_t)B_SZ*K_DIR*L_SZ*XDBL_STRIDE) // (B,K,DE,L) scan output
#define OFF_YNZ  (OFF_YS   + (size_t)B_SZ*K_DIR*DE_SZ*L_SZ)   // (B*L, DE)   normalized * gated

__device__ __forceinline__ int phys_idx(int k, int l) {
  // map scan index l -> row-major spatial index in (H,W), per direction k
  if (k == 0) return l;
  if (k == 1) return L_SZ - 1 - l;
  if (k == 2) return ((l & 63) << 6) | (l >> 6);        // l = w*H + h -> h*W + w
  int m = L_SZ - 1 - l;
  return ((m & 63) << 6) | (m >> 6);
}

__device__ __forceinline__ float silu_f(float v) {
  return v / (1.f + __expf(-v));
}

// ============ K1: in_proj GEMM (8192x96 @ 96x384), split xp / silu(z) ============
__global__ void k_inproj(const float* __restrict__ x,
                         const float* __restrict__ w,   // (384, 96)
                         float* __restrict__ ws) {
  const int wave = blockIdx.x * (blockDim.x >> 5) + (threadIdx.x >> 5); // 0..12287
  const int lane = threadIdx.x & 31;
  const int lr = lane & 15, hi = lane >> 4;
  const int tm = wave & 511;          // 512 M-tiles
  const int tn = wave >> 9;           // 24  N-tiles
  const int m0 = tm << 4, n0 = tn << 4;
  v8f acc = {};
#if HAVE_WMMA_F32
  const float* arow = x + (size_t)(m0 + lr) * DM_SZ;
  const float* brow = w + (size_t)(n0 + lr) * DM_SZ;   // B(k,n) = w[n*96+k]
  for (int kk = 0; kk < DM_SZ; kk += 4) {
    v2f a, b;
    a.x = arow[kk + 2*hi]; a.y = arow[kk + 2*hi + 1];
    b.x = brow[kk + 2*hi]; b.y = brow[kk + 2*hi + 1];
    acc = __builtin_amdgcn_wmma_f32_16x16x4_f32(false, a, false, b,
                                                (short)0, acc, false, false);
  }
#else
  const float* bcol = w + (size_t)(n0 + lr) * DM_SZ;
  for (int kk = 0; kk < DM_SZ; ++kk) {
    float bv = bcol[kk];
    #pragma unroll
    for (int j = 0; j < 8; ++j)
      acc[j] += x[(size_t)(m0 + j + 8*hi) * DM_SZ + kk] * bv;
  }
#endif
  #pragma unroll
  for (int j = 0; j < 8; ++j) {
    int gr = m0 + j + 8*hi;           // global row in [0, B*L)
    int n  = n0 + lr;                 // output column in [0, 384)
    float v = acc[j];
    int b = gr >> 12, l = gr & (L_SZ - 1);
    if (n < DE_SZ) {
      ws[OFF_XC0 + ((size_t)(b*DE_SZ + n)) * L_SZ + l] = v;          // xp, channel-major
    } else {
      ws[OFF_ZS + (size_t)gr * DE_SZ + (n - DE_SZ)] = silu_f(v);     // silu(z), row-major
    }
  }
}

// ============ K2: depthwise 3x3 conv + bias + SiLU ============
__global__ void k_conv(const float* __restrict__ cw,   // (DE,1,3,3)
                       const float* __restrict__ cb,   // (DE,)
                       float* __restrict__ ws) {
  int idx = blockIdx.x * blockDim.x + threadIdx.x;     // B*DE*L threads
  int p  = idx & (L_SZ - 1);
  int bc = idx >> 12;
  int c  = bc % DE_SZ;
  int b  = bc / DE_SZ;
  int h = p >> 6, w = p & 63;
  const float* src = ws + OFF_XC0 + (size_t)(b*DE_SZ + c) * L_SZ;
  const float* kw  = cw + c * 9;
  float acc = cb[c];
  #pragma unroll
  for (int i = 0; i < 3; ++i) {
    int hh = h + i - 1;
    if (hh < 0 || hh >= H_SZ) continue;
    #pragma unroll
    for (int j = 0; j < 3; ++j) {
      int wp = w + j - 1;
      if (wp < 0 || wp >= W_SZ) continue;
      acc += src[hh * W_SZ + wp] * kw[i*3 + j];
    }
  }
  ws[OFF_XC1 + (size_t)(b*DE_SZ + c) * L_SZ + p] = silu_f(acc);
}

// ============ K3: x_dbl = x_proj_weight[k] (38x192) @ xs[b,k] (192xL) ============
// Output is l-major with padded stride 40 so the scan can issue b128 loads.
__global__ void k_xdbl(const float* __restrict__ xpw,  // (K, 38, 192)
                       float* __restrict__ ws) {
  const int bk = blockIdx.y;                           // 0..7
  const int b = bk >> 2, k = bk & 3;
  const int wave = blockIdx.x * (blockDim.x >> 5) + (threadIdx.x >> 5); // 0..767
  const int lane = threadIdx.x & 31, lr = lane & 15, hi = lane >> 4;
  const int tn = wave & 255, tm = wave >> 8;           // 256 N-tiles, 3 M-tiles
  const int m0 = tm << 4;
  const int l  = (tn << 4) + lr;                       // scan position (this lane's N column)
  const int pl = phys_idx(k, l);
  const float* xc1 = ws + OFF_XC1 + (size_t)b * DE_SZ * L_SZ;
  v8f acc = {};
#if HAVE_WMMA_F32
  const int  c     = m0 + lr;
  const bool valid = (c < CPROJ);
  const float* apw = xpw + ((size_t)k * CPROJ + (valid ? c : 0)) * DE_SZ;
  for (int kk = 0; kk < DE_SZ; kk += 4) {
    int d = kk + 2*hi;
    v2f a, bfr;
    a.x = valid ? apw[d]     : 0.f;
    a.y = valid ? apw[d + 1] : 0.f;
    bfr.x = xc1[(size_t)d       * L_SZ + pl];
    bfr.y = xc1[(size_t)(d + 1) * L_SZ + pl];
    acc = __builtin_amdgcn_wmma_f32_16x16x4_f32(false, a, false, bfr,
                                                (short)0, acc, false, false);
  }
#else
  for (int kk = 0; kk < DE_SZ; ++kk) {
    float bv = xc1[(size_t)kk * L_SZ + pl];
    #pragma unroll
    for (int j = 0; j < 8; ++j) {
      int cj = m0 + j + 8*hi;
      float av = (cj < CPROJ) ? xpw[((size_t)k*CPROJ + cj)*DE_SZ + kk] : 0.f;
      acc[j] += av * bv;
    }
  }
#endif
  float* out = ws + OFF_XDBL + (size_t)bk * L_SZ * XDBL_STRIDE;
  #pragma unroll
  for (int j = 0; j < 8; ++j) {
    int cj = m0 + j + 8*hi;
    if (cj < CPROJ) {
      int slot = (cj < R_SZ) ? cj : cj + 2;   // dt: 0..5, B: 8..23, C: 24..39
      out[(size_t)l * XDBL_STRIDE + slot] = acc[j];
    }
  }
}

// ============ K4: selective scan; one lane per channel d, h[16] in regs ============
struct StepRow {
  float4 t0; float2 t1;        // dt projection inputs (6)
  float4 B0, B1, B2, B3;       // B (16)
  float4 C0, C1, C2, C3;       // C (16)
};

__device__ __forceinline__ void load_row(const float* __restrict__ row, StepRow& r) {
  r.t0 = *reinterpret_cast<const float4*>(row);
  r.t1 = *reinterpret_cast<const float2*>(row + 4);
  r.B0 = *reinterpret_cast<const float4*>(row + 8);
  r.B1 = *reinterpret_cast<const float4*>(row + 12);
  r.B2 = *reinterpret_cast<const float4*>(row + 16);
  r.B3 = *reinterpret_cast<const float4*>(row + 20);
  r.C0 = *reinterpret_cast<const float4*>(row + 24);
  r.C1 = *reinterpret_cast<const float4*>(row + 28);
  r.C2 = *reinterpret_cast<const float4*>(row + 32);
  r.C3 = *reinterpret_cast<const float4*>(row + 36);
}

__device__ __forceinline__ void step4(float delta, float du,
                                      const float* __restrict__ Ar,
                                      float* __restrict__ h,
                                      float4 Bv, float4 Cv, float& y) {
  float hv;
  hv = __expf(delta * Ar[0]) * h[0] + du * Bv.x; h[0] = hv; y += hv * Cv.x;
  hv = __expf(delta * Ar[1]) * h[1] + du * Bv.y; h[1] = hv; y += hv * Cv.y;
  hv = __expf(delta * Ar[2]) * h[2] + du * Bv.z; h[2] = hv; y += hv * Cv.z;
  hv = __expf(delta * Ar[3]) * h[3] + du * Bv.w; h[3] = hv; y += hv * Cv.w;
}

__global__ void k_scan(const float* __restrict__ A_logs, // (K*DE, N)
                       const float* __restrict__ dtw,    // (K, DE, R)
                       const float* __restrict__ dtb,    // (K, DE)
                       const float* __restrict__ Dsp,    // (K*DE,)
                       float* __restrict__ ws) {
  const int bk = blockIdx.x;                      // 0..7
  const int b = bk >> 2, k = bk & 3;
  const int d = blockIdx.y * 32 + threadIdx.x;    // 0..191, one wave per block
  const int kd = k * DE_SZ + d;
  float Arow[N_ST];
  #pragma unroll
  for (int n = 0; n < N_ST; ++n) Arow[n] = -__expf(A_logs[(size_t)kd * N_ST + n]);
  float w6[R_SZ];
  #pragma unroll
  for (int r = 0; r < R_SZ; ++r) w6[r] = dtw[(size_t)kd * R_SZ + r];
  const float bias = dtb[kd];
  const float dsv  = Dsp[kd];
  const float* xd = ws + OFF_XDBL + (size_t)bk * L_SZ * XDBL_STRIDE;
  const float* ub = ws + OFF_XC1  + ((size_t)b * DE_SZ + d) * L_SZ;
  float* ys       = ws + OFF_YS   + ((size_t)bk * DE_SZ + d) * L_SZ;
  float h[N_ST];
  #pragma unroll
  for (int n = 0; n < N_ST; ++n) h[n] = 0.f;

  StepRow cur, nxt;
  load_row(xd, cur);
  float u_cur = ub[phys_idx(k, 0)];
  for (int l = 0; l < L_SZ; ++l) {
    // software pipeline: issue next step's loads before computing this step.
    // (row L_SZ / index L_SZ reads land inside the workspace; values unused.)
    const float* nrow = xd + (size_t)(l + 1) * XDBL_STRIDE;
    load_row(nrow, nxt);
    float u_nxt = ub[phys_idx(k, l + 1)];
    __builtin_prefetch(nrow + 7 * XDBL_STRIDE, 0, 1);   // global_prefetch_b8

    float dtr = bias
              + w6[0]*cur.t0.x + w6[1]*cur.t0.y + w6[2]*cur.t0.z
              + w6[3]*cur.t0.w + w6[4]*cur.t1.x + w6[5]*cur.t1.y;
    float delta = (dtr > 20.f) ? dtr : log1pf(__expf(dtr));   // softplus
    float du = delta * u_cur;
    float y  = u_cur * dsv;
    step4(delta, du, Arow + 0,  h + 0,  cur.B0, cur.C0, y);
    step4(delta, du, Arow + 4,  h + 4,  cur.B1, cur.C1, y);
    step4(delta, du, Arow + 8,  h + 8,  cur.B2, cur.C2, y);
    step4(delta, du, Arow + 12, h + 12, cur.B3, cur.C3, y);
    ys[l] = y;

    cur = nxt; u_cur = u_nxt;
  }
}

// ============ K5a: merge 4 directions + LayerNorm + gate with silu(z) ============
__global__ void k_combine(const float* __restrict__ gamma,
                          const float* __restrict__ beta,
                          float* __restrict__ ws) {
  const int row = blockIdx.x;                // b*L + p
  const int b = row >> 12;
  const int p = row & (L_SZ - 1);
  const int t = threadIdx.x;                 // 64 threads, 3 channels each
  const int l2 = ((p & 63) << 6) | (p >> 6);
  const float* ysb = ws + OFF_YS + (size_t)b * K_DIR * DE_SZ * L_SZ;
  float yv[3];
  float s = 0.f, sq = 0.f;
  #pragma unroll
  for (int q = 0; q < 3; ++q) {
    int d = t + 64*q;
    float y = ysb[(size_t)(0*DE_SZ + d) * L_SZ + p]
            + ysb[(size_t)(1*DE_SZ + d) * L_SZ + (L_SZ - 1 - p)]
            + ysb[(size_t)(2*DE_SZ + d) * L_SZ + l2]
            + ysb[(size_t)(3*DE_SZ + d) * L_SZ + (L_SZ - 1 - l2)];
    yv[q] = y; s += y; sq += y * y;
  }
  __shared__ float s1[64], s2[64];
  __shared__ float mu_s, ri_s;
  s1[t] = s; s2[t] = sq;
  __syncthreads();
  for (int st = 32; st > 0; st >>= 1) {
    if (t < st) { s1[t] += s1[t + st]; s2[t] += s2[t + st]; }
    __syncthreads();
  }
  if (t == 0) {
    float mu  = s1[0] * (1.f / DE_SZ);
    float var = s2[0] * (1.f / DE_SZ) - mu * mu;
    mu_s = mu; ri_s = rsqrtf(var + 1e-5f);
  }
  __syncthreads();
  const float mu = mu_s, ri = ri_s;
  #pragma unroll
  for (int q = 0; q < 3; ++q) {
    int d = t + 64*q;
    float nv = (yv[q] - mu) * ri * gamma[d] + beta[d];
    float z  = ws[OFF_ZS + (size_t)row * DE_SZ + d];
    ws[OFF_YNZ + (size_t)row * DE_SZ + d] = nv * z;
  }
}

// ============ K5b: out_proj GEMM (8192x192 @ 192x96) -> d_out ============
__global__ void k_outproj(const float* __restrict__ opw,  // (96, 192)
                          const float* __restrict__ ws,
                          float* __restrict__ out) {
  const int wave = blockIdx.x * (blockDim.x >> 5) + (threadIdx.x >> 5); // 0..3071
  const int lane = threadIdx.x & 31, lr = lane & 15, hi = lane >> 4;
  const int tm = wave & 511, tn = wave >> 9;   // 512 M-tiles, 6 N-tiles
  const int m0 = tm << 4, n0 = tn << 4;
  v8f acc = {};
#if HAVE_WMMA_F32
  const float* arow = ws + OFF_YNZ + (size_t)(m0 + lr) * DE_SZ;
  const float* brow = opw + (size_t)(n0 + lr) * DE_SZ;   // B(k,n) = opw[n*192+k]
  for (int kk = 0; kk < DE_SZ; kk += 4) {
    v2f a, bfr;
    a.x = arow[kk + 2*hi];   a.y = arow[kk + 2*hi + 1];
    bfr.x = brow[kk + 2*hi]; bfr.y = brow[kk + 2*hi + 1];
    acc = __builtin_amdgcn_wmma_f32_16x16x4_f32(false, a, false, bfr,
                                                (short)0, acc, false, false);
  }
#else
  for (int kk = 0; kk < DE_SZ; ++kk) {
    float bv = opw[(size_t)(n0 + lr) * DE_SZ + kk];
    #pragma unroll
    for (int j = 0; j < 8; ++j)
      acc[j] += ws[OFF_YNZ + (size_t)(m0 + j + 8*hi) * DE_SZ + kk] * bv;
  }
#endif
  #pragma unroll
  for (int j = 0; j < 8; ++j) {
    int gr = m0 + j + 8*hi;
    int n  = n0 + lr;
    out[(size_t)gr * DM_SZ + n] = acc[j];
  }
}

extern "C" void kernel_launch(void* const* d_in, const int* in_sizes, int n_in,
                              void* d_out, int out_size, void* d_ws, size_t ws_size,
                              hipStream_t stream) {
  (void)in_sizes; (void)n_in; (void)out_size; (void)ws_size;
  const float* x    = (const float*)d_in[0];
  const float* ipw  = (const float*)d_in[1];
  const float* cw   = (const float*)d_in[2];
  const float* cb   = (const float*)d_in[3];
  const float* xpw  = (const float*)d_in[4];
  const float* dtw  = (const float*)d_in[5];
  const float* dtb  = (const float*)d_in[6];
  const float* alog = (const float*)d_in[7];
  const float* Dsp  = (const float*)d_in[8];
  const float* gam  = (const float*)d_in[9];
  const float* bet  = (const float*)d_in[10];
  const float* opw  = (const float*)d_in[11];
  float* out = (float*)d_out;
  float* ws  = (float*)d_ws;

  // 1) in_proj: 512*24 = 12288 wave-tiles, 4 waves/block
  k_inproj<<<3072, 128, 0, stream>>>(x, ipw, ws);
  // 2) depthwise conv: B*DE*L threads
  k_conv<<<(B_SZ * DE_SZ * L_SZ) / 256, 256, 0, stream>>>(cw, cb, ws);
  // 3) x_dbl per (b,k): 3*256 = 768 wave-tiles each
  k_xdbl<<<dim3(192, B_SZ * K_DIR), 128, 0, stream>>>(xpw, ws);
  // 4) selective scan: one wave per block, spread across 48 WGPs
  k_scan<<<dim3(B_SZ * K_DIR, DE_SZ / 32), 32, 0, stream>>>(alog, dtw, dtb, Dsp, ws);
  // 5) merge + LN + gate: one block per (b, spatial)
  k_combine<<<B_SZ * L_SZ, 64, 0, stream>>>(gam, bet, ws);
  // 6) out_proj: 512*6 = 3072 wave-tiles
  k_outproj<<<768, 128, 0, stream>>>(opw, ws, out);
}